// Tokenizer_50955491999984
// MI455X (gfx1250) — compile-verified
//
#include <hip/hip_runtime.h>
#include <hip/hip_bf16.h>

// ---------------------------------------------------------------------------
// GIN forward: 5 x { agg = scatter_add(h[src]->dst) + 0.5*h ; batchnorm }
// h: N x 64 f32 (lives in d_out, rewritten every layer)
// agg + stats scratch live in d_ws.
// Column mean/var reduction uses V_WMMA_F32_16X16X4_F32 with A = ones.
// ---------------------------------------------------------------------------

typedef float v2f __attribute__((ext_vector_type(2)));
typedef float v8f __attribute__((ext_vector_type(8)));

#define EMB_DIM 64
#define GIN_EPS 0.5f
#define BN_EPS 1e-5f

// h[node] = embedding_table[x[node,0]]
__global__ void gin_embed(const int* __restrict__ x,
                          const float* __restrict__ emb,
                          float* __restrict__ h, int nNodes) {
    int tid = blockIdx.x * blockDim.x + threadIdx.x;
    int node = tid >> 4;                 // 16 threads / node
    if (node >= nNodes) return;
    int c0 = (tid & 15) * 4;
    int a = x[(size_t)node * 2];         // atom type (column 0)
    float4 v = *reinterpret_cast<const float4*>(emb + (size_t)a * EMB_DIM + c0);
    *reinterpret_cast<float4*>(h + (size_t)node * EMB_DIM + c0) = v;
}

// agg = GIN_EPS * h ; also zero the 128-float stats buffer
__global__ void gin_init(const float* __restrict__ h,
                         float* __restrict__ agg,
                         float* __restrict__ stats, long total) {
    long i = (long)blockIdx.x * blockDim.x + threadIdx.x;
    if (i < total) agg[i] = GIN_EPS * h[i];
    if (blockIdx.x == 0 && threadIdx.x < 128) stats[threadIdx.x] = 0.0f;
}

// agg[dst] += h[src]  (16 threads per edge, float4 per thread, f32 atomics)
__global__ void gin_scatter(const float* __restrict__ h,
                            const int* __restrict__ src,
                            const int* __restrict__ dst,
                            float* __restrict__ agg, int nEdges) {
    int tid = blockIdx.x * blockDim.x + threadIdx.x;
    int c0 = (tid & 15) * 4;
    int e = tid >> 4;
    int stride = (gridDim.x * blockDim.x) >> 4;
    for (; e < nEdges; e += stride) {
        int s = src[e];
        int d = dst[e];
        float4 v = *reinterpret_cast<const float4*>(h + (size_t)s * EMB_DIM + c0);
        float* out = agg + (size_t)d * EMB_DIM + c0;
        atomicAdd(out + 0, v.x);
        atomicAdd(out + 1, v.y);
        atomicAdd(out + 2, v.z);
        atomicAdd(out + 3, v.w);
    }
}

// Column sums + column sums-of-squares of agg (nRows x 64) via WMMA.
// A = ones(16x4)  =>  D[m][n] = sum_k B[k][n] + C[m][n]  (every row of D equal)
// Each wave owns one 16-column group (gw & 3) and grid-strides over 4-row
// chunks with a guard-free inner loop; the (nRows & 3) remainder is folded in
// by four designated waves using plain scalar adds.
__global__ void gin_stats(const float* __restrict__ agg,
                          float* __restrict__ stats, int nRows) {
    const int lane = threadIdx.x & 31;
    const int wavesPerBlock = blockDim.x >> 5;
    const int gw = blockIdx.x * wavesPerBlock + (threadIdx.x >> 5);
    const int g = gw & 3;                 // column group 0..3
    const int half = lane >> 4;           // 0 or 1
    const int col = g * 16 + (lane & 15); // feature column this lane reduces
    const int nFull = nRows >> 2;         // full 4-row chunks
    const int chunkStride = (gridDim.x * wavesPerBlock) >> 2;

    v2f a; a.x = 1.0f; a.y = 1.0f;        // all-ones A matrix
    v8f csum = {};
    v8f csq = {};

    // Guard-free hot loop: lane supplies rows (4c+half) and (4c+2+half).
    const float* p = agg + (size_t)(((gw >> 2) * 4) + half) * EMB_DIM + col;
    const size_t step = (size_t)chunkStride * 4 * EMB_DIM;
#pragma unroll 2
    for (int chunk = gw >> 2; chunk < nFull; chunk += chunkStride, p += step) {
        v2f b;
        b.x = p[0];
        b.y = p[2 * EMB_DIM];
        v2f b2;
        b2.x = b.x * b.x;
        b2.y = b.y * b.y;
        csum = __builtin_amdgcn_wmma_f32_16x16x4_f32(
            false, a, false, b, (short)0, csum, false, false);
        csq = __builtin_amdgcn_wmma_f32_16x16x4_f32(
            false, a, false, b2, (short)0, csq, false, false);
    }

    // Remainder rows (empty when nRows % 4 == 0): one wave per column group.
    float tailS = 0.0f, tailQ = 0.0f;
    if (gw < 4) {
        for (int r = nFull * 4; r < nRows; ++r) {
            float v = agg[(size_t)r * EMB_DIM + col];
            tailS += v;
            tailQ += v * v;
        }
    }

    // D row 0, lanes 0..15 hold colsum[col]; fold partials into stats.
    if (lane < 16) {
        atomicAdd(&stats[col], csum[0] + tailS);
        atomicAdd(&stats[64 + col], csq[0] + tailQ);
    }
}

// h = (agg - mean) * rsqrt(var + eps),  mean/var from stats sums
__global__ void gin_norm(const float* __restrict__ agg,
                         const float* __restrict__ stats,
                         float* __restrict__ h, int nNodes) {
    int tid = blockIdx.x * blockDim.x + threadIdx.x;
    int node = tid >> 4;
    if (node >= nNodes) return;
    int c0 = (tid & 15) * 4;
    float invN = 1.0f / (float)nNodes;

    float4 v = *reinterpret_cast<const float4*>(agg + (size_t)node * EMB_DIM + c0);
    float4 s = *reinterpret_cast<const float4*>(stats + c0);
    float4 q = *reinterpret_cast<const float4*>(stats + 64 + c0);

    float m0 = s.x * invN, m1 = s.y * invN, m2 = s.z * invN, m3 = s.w * invN;
    float i0 = 1.0f / sqrtf(q.x * invN - m0 * m0 + BN_EPS);
    float i1 = 1.0f / sqrtf(q.y * invN - m1 * m1 + BN_EPS);
    float i2 = 1.0f / sqrtf(q.z * invN - m2 * m2 + BN_EPS);
    float i3 = 1.0f / sqrtf(q.w * invN - m3 * m3 + BN_EPS);

    float4 o;
    o.x = (v.x - m0) * i0;
    o.y = (v.y - m1) * i1;
    o.z = (v.z - m2) * i2;
    o.w = (v.w - m3) * i3;
    *reinterpret_cast<float4*>(h + (size_t)node * EMB_DIM + c0) = o;
}

extern "C" void kernel_launch(void* const* d_in, const int* in_sizes, int n_in,
                              void* d_out, int out_size, void* d_ws, size_t ws_size,
                              hipStream_t stream) {
    (void)n_in; (void)out_size; (void)ws_size;
    const int* x = (const int*)d_in[0];            // (N, 2) int
    const int* edge_index = (const int*)d_in[1];   // (2, E) int
    const float* emb = (const float*)d_in[2];      // (120, 64) f32

    const int nNodes = in_sizes[0] / 2;
    const int nEdges = in_sizes[1] / 2;
    const int* src = edge_index;                   // row 0
    const int* dst = edge_index + nEdges;          // row 1

    float* h = (float*)d_out;                      // N x 64, in-place across layers
    float* stats = (float*)d_ws;                   // 128 floats (sum | sumsq)
    float* agg = stats + 128;                      // N x 64 scratch

    const long total = (long)nNodes * EMB_DIM;
    const long vec = (long)nNodes * 16;            // float4-granule threads

    {
        int blocks = (int)((vec + 255) / 256);
        gin_embed<<<blocks, 256, 0, stream>>>(x, emb, h, nNodes);
    }
    for (int layer = 0; layer < 5; ++layer) {
        {
            int blocks = (int)((total + 255) / 256);
            gin_init<<<blocks, 256, 0, stream>>>(h, agg, stats, total);
        }
        gin_scatter<<<4096, 256, 0, stream>>>(h, src, dst, agg, nEdges);
        gin_stats<<<256, 256, 0, stream>>>(agg, stats, nNodes);
        {
            int blocks = (int)((vec + 255) / 256);
            gin_norm<<<blocks, 256, 0, stream>>>(agg, stats, h, nNodes);
        }
    }
}